// LossImprovedAOE_ROE_38336878084355
// MI455X (gfx1250) — compile-verified
//
#include <hip/hip_runtime.h>

// ---------------------------------------------------------------------------
// LossImprovedAOE_ROE for MI455X (gfx1250, wave32)
//   xs, hat_xs : (16, 65536, 3) fp32    ->  scalar fp32 loss
//
// Roofline: 25.2 MB input @ 23.3 TB/s = ~1.1 us floor. Stage 1 keeps every
// intermediate 3x3 in VGPRs, uses polynomial sin(a)/a & (1-cos a)/a^2 for the
// 16 small-angle (a <= ~0.1 rad, DT=0.01) chain rotations (err ~3e-14, beats
// fp32 trig), so only one full-range sinf/cosf + ~1.5 acosf per thread remain.
// Level-5 pairing via wave32 shuffles; deterministic block partials to d_ws.
// Stage 2: single wave; V_WMMA_F32_16X16X4_F32 ones-matmul column-sum
// reduction (layout-independent: sum of all C elements == 16 * total).
// ---------------------------------------------------------------------------

typedef __attribute__((ext_vector_type(2))) float v2f;
typedef __attribute__((ext_vector_type(8))) float v8f;

#define NGROUPS   65536          // 16*65536/16  level-4 elements
#define AOE_CNT   65536.0f
#define HUBER_CNT 98064.0f       // 16 * (2048-5) * 3
#define INV_HUBER 200.0f         // 1/0.005

struct M3 { float m[9]; };

__device__ __forceinline__ M3 m3mul(const M3& A, const M3& B) {
    M3 C;
#pragma unroll
    for (int i = 0; i < 3; ++i)
#pragma unroll
        for (int j = 0; j < 3; ++j)
            C.m[i*3+j] = A.m[i*3+0]*B.m[0*3+j]
                       + A.m[i*3+1]*B.m[1*3+j]
                       + A.m[i*3+2]*B.m[2*3+j];
    return C;
}

// A^T * B  (einsum 'nji,njk->nik')
__device__ __forceinline__ M3 m3Tmul(const M3& A, const M3& B) {
    M3 C;
#pragma unroll
    for (int i = 0; i < 3; ++i)
#pragma unroll
        for (int j = 0; j < 3; ++j)
            C.m[i*3+j] = A.m[0*3+i]*B.m[0*3+j]
                       + A.m[1*3+i]*B.m[1*3+j]
                       + A.m[2*3+i]*B.m[2*3+j];
    return C;
}

__device__ __forceinline__ M3 buildR(float x, float y, float z,
                                     float s, float c, float a2) {
    // R = I + s*K + c*(phi phi^T - a2 I)
    M3 R;
    R.m[0] = 1.0f + c*(x*x - a2); R.m[1] = -s*z + c*x*y;        R.m[2] =  s*y + c*x*z;
    R.m[3] =  s*z + c*x*y;        R.m[4] = 1.0f + c*(y*y - a2); R.m[5] = -s*x + c*y*z;
    R.m[6] = -s*y + c*x*z;        R.m[7] =  s*x + c*y*z;        R.m[8] = 1.0f + c*(z*z - a2);
    return R;
}

// Small-angle Rodrigues (a2 <= ~0.01 here): 3-term Horner for sin(a)/a and
// (1-cos a)/a^2; truncation ~a^8/9! — below fp32 ulp for this input range.
__device__ __forceinline__ M3 so3exp_small(float x, float y, float z) {
    float a2 = x*x + y*y + z*z;
    float s = 1.0f - (a2*(1.0f/6.0f))  * (1.0f - (a2*(1.0f/20.0f)) * (1.0f - a2*(1.0f/42.0f)));
    float c = 0.5f * (1.0f - (a2*(1.0f/12.0f)) * (1.0f - (a2*(1.0f/30.0f)) * (1.0f - a2*(1.0f/56.0f))));
    return buildR(x, y, z, s, c, a2);
}

// Full-range Rodrigues matching the reference's small-angle switch.
__device__ __forceinline__ M3 so3exp(float x, float y, float z) {
    float a2  = x*x + y*y + z*z;
    float a2s = fmaxf(a2, 1e-16f);
    float a   = sqrtf(a2s);
    bool  sm  = a2 < 1e-8f;
    float s   = sm ? (1.0f - a2 * (1.0f/6.0f))  : (sinf(a) / a);
    float c   = sm ? (0.5f - a2 * (1.0f/24.0f)) : ((1.0f - cosf(a)) / a2s);
    return buildR(x, y, z, s, c, a2);
}

__device__ __forceinline__ float huber_elem(float x) {
    float ax = fabsf(x);
    return (ax < 1.0f) ? 0.5f*x*x : ax - 0.5f;
}

__global__ __launch_bounds__(256)
void loss_stage1(const float* __restrict__ xs, const float* __restrict__ hat,
                 float* __restrict__ part) {
    const int g = blockIdx.x * 256 + threadIdx.x;       // group id, < 65536

    const float* xp = xs + (size_t)g * 48;              // xs[:, ::16] element g
    __builtin_prefetch(xp, 0, 0);                       // global_prefetch_b8

    const float4* hv = reinterpret_cast<const float4*>(hat + (size_t)g * 48);

    // ---- chain product of 16 so3_exp(0.01*hat), strict left->right order.
    // Peel step 0 so we never multiply by identity.
    M3 Om;
    {
        float4 q0 = hv[0], q1 = hv[1], q2 = hv[2];
        float f[12] = {q0.x,q0.y,q0.z,q0.w, q1.x,q1.y,q1.z,q1.w,
                       q2.x,q2.y,q2.z,q2.w};
        Om = so3exp_small(f[0]*0.01f, f[1]*0.01f, f[2]*0.01f);
#pragma unroll
        for (int t = 1; t < 4; ++t) {
            M3 O = so3exp_small(f[3*t]*0.01f, f[3*t+1]*0.01f, f[3*t+2]*0.01f);
            Om = m3mul(Om, O);
        }
    }
#pragma unroll
    for (int c2 = 1; c2 < 4; ++c2) {
        float4 q0 = hv[c2*3+0], q1 = hv[c2*3+1], q2 = hv[c2*3+2];
        float f[12] = {q0.x,q0.y,q0.z,q0.w, q1.x,q1.y,q1.z,q1.w,
                       q2.x,q2.y,q2.z,q2.w};
#pragma unroll
        for (int t = 0; t < 4; ++t) {
            M3 O = so3exp_small(f[3*t]*0.01f, f[3*t+1]*0.01f, f[3*t+2]*0.01f);
            Om = m3mul(Om, O);
        }
    }

    // ---- base-level ground-truth rotation (full-range trig, once/thread)
    M3 X = so3exp(xp[0], xp[1], xp[2]);

    // ---- AOE: angle(Om^T X)^2
    M3 rel = m3Tmul(Om, X);
    float tr = rel.m[0] + rel.m[4] + rel.m[8];
    float ca = fminf(fmaxf((tr - 1.0f) * 0.5f, -1.0f), 1.0f);
    float ang = acosf(ca);
    float aSum = ang * ang;

    // ---- level-5: pair adjacent lanes (wave32 shuffles), even lanes compute
    M3 OmP, XP;
#pragma unroll
    for (int k = 0; k < 9; ++k) {
        OmP.m[k] = __shfl_down(Om.m[k], 1, 32);
        XP.m[k]  = __shfl_down(X.m[k],  1, 32);
    }
    float hSum = 0.0f;
    if ((threadIdx.x & 1) == 0) {
        int i5 = g >> 1;
        if ((i5 & 2047) >= 5) {                          // [:, N0:] mask
            M3 Om5 = m3mul(Om, OmP);
            M3 X5  = m3mul(X,  XP);
            M3 r5  = m3Tmul(Om5, X5);
            float tr5 = r5.m[0] + r5.m[4] + r5.m[8];
            float ca5 = fminf(fmaxf((tr5 - 1.0f)*0.5f, -1.0f), 1.0f);
            float a5  = acosf(ca5);
            float sa5 = sinf(a5);
            float fac = (sa5 < 1e-6f) ? 0.5f : (a5 / (2.0f * sa5));
            float w0 = (r5.m[7] - r5.m[5]) * fac;
            float w1 = (r5.m[2] - r5.m[6]) * fac;
            float w2 = (r5.m[3] - r5.m[1]) * fac;
            hSum = huber_elem(w0*INV_HUBER) + huber_elem(w1*INV_HUBER)
                 + huber_elem(w2*INV_HUBER);
        }
    }

    // ---- deterministic block reduction
#pragma unroll
    for (int off = 16; off > 0; off >>= 1) {
        aSum += __shfl_down(aSum, off, 32);
        hSum += __shfl_down(hSum, off, 32);
    }
    __shared__ float redA[8], redH[8];
    int wid = threadIdx.x >> 5, lane = threadIdx.x & 31;
    if (lane == 0) { redA[wid] = aSum; redH[wid] = hSum; }
    __syncthreads();
    if (threadIdx.x == 0) {
        float a = 0.0f, h = 0.0f;
#pragma unroll
        for (int w = 0; w < 8; ++w) { a += redA[w]; h += redH[w]; }
        part[blockIdx.x]       = a;
        part[256 + blockIdx.x] = h;
    }
}

// Stage 2: one wave. A = ones(16x4) -> C[m,n] = sum_k B[k,n]; every row of C
// equals the column sums, so sum(all C)/16 == sum(all B) independent of the
// B lane layout. 4 accumulating WMMAs consume 256 values per accumulator.
__global__ __launch_bounds__(32)
void loss_stage2(const float* __restrict__ part, float* __restrict__ out) {
    const int lane = threadIdx.x;                       // 0..31, EXEC all ones
    v2f a; a.x = 1.0f; a.y = 1.0f;
    v8f cA = {}, cH = {};
#pragma unroll
    for (int i = 0; i < 4; ++i) {
        v2f bA, bH;
        bA.x = part[i*64 + lane*2];       bA.y = part[i*64 + lane*2 + 1];
        bH.x = part[256 + i*64 + lane*2]; bH.y = part[256 + i*64 + lane*2 + 1];
        cA = __builtin_amdgcn_wmma_f32_16x16x4_f32(false, a, false, bA,
                                                   (short)0, cA, false, false);
        cH = __builtin_amdgcn_wmma_f32_16x16x4_f32(false, a, false, bH,
                                                   (short)0, cH, false, false);
    }
    float sA = 0.0f, sH = 0.0f;
#pragma unroll
    for (int v = 0; v < 8; ++v) { sA += cA[v]; sH += cH[v]; }
#pragma unroll
    for (int off = 16; off > 0; off >>= 1) {
        sA += __shfl_down(sA, off, 32);
        sH += __shfl_down(sH, off, 32);
    }
    if (lane == 0) {
        sA *= (1.0f/16.0f);                             // 16 identical rows
        sH *= (1.0f/16.0f);
        // 0.2*mean(angle^2) + 0.1 * (W*HUBER^2 * mean_huber) / 2
        out[0] = 0.2f * (sA / AOE_CNT) + 1.25f * (sH / HUBER_CNT);
    }
}

extern "C" void kernel_launch(void* const* d_in, const int* in_sizes, int n_in,
                              void* d_out, int out_size, void* d_ws, size_t ws_size,
                              hipStream_t stream) {
    const float* xs  = (const float*)d_in[0];
    const float* hat = (const float*)d_in[1];
    float* ws  = (float*)d_ws;       // 512 floats of partials
    float* out = (float*)d_out;

    loss_stage1<<<NGROUPS / 256, 256, 0, stream>>>(xs, hat, ws);
    loss_stage2<<<1, 32, 0, stream>>>(ws, out);
}